// RegisterTiledHyperbolicDistance_16870631539337
// MI455X (gfx1250) — compile-verified
//
#include <hip/hip_runtime.h>
#include <hip/hip_bf16.h>

typedef float v2f __attribute__((ext_vector_type(2)));
typedef float v4f __attribute__((ext_vector_type(4)));
typedef float v8f __attribute__((ext_vector_type(8)));

// Problem dims from the reference (fixed by the harness).
constexpr int Bb = 4, Hh = 8, Nn = 2048, Dd = 64;

// Tiling.
constexpr int WG_M = 64;          // workgroup M tile (2 waves of 32)
constexpr int WG_N = 128;         // workgroup N tile (4 waves of 32)
constexpr int LDS_STRIDE = 68;    // 64 + 4 pad floats -> 272B rows: 16B aligned, no bank conflicts

__global__ __launch_bounds__(256)
void hyperdist_wmma_kernel(const float* __restrict__ q,
                           const float* __restrict__ k,
                           float* __restrict__ out) {
  __shared__ float q_s[WG_M * LDS_STRIDE];   // 64 x 64 tile of q rows
  __shared__ float k_s[WG_N * LDS_STRIDE];   // 128 x 64 tile of k rows
  __shared__ float qn_s[WG_M];               // ||q_row||^2
  __shared__ float kn_s[WG_N];               // ||k_row||^2

  const int tid    = threadIdx.x;
  const int bh     = blockIdx.z;             // fused batch*head, 0..31
  const int tile_m = blockIdx.y * WG_M;
  const int tile_n = blockIdx.x * WG_N;

  const float* __restrict__ qb = q + (size_t)bh * Nn * Dd;
  const float* __restrict__ kb = k + (size_t)bh * Nn * Dd;

  // ---- stage q tile: 64 rows x 64 cols = 1024 float4, 4 per thread ----
#pragma unroll
  for (int i = 0; i < 4; ++i) {
    int idx = tid + i * 256;
    int r   = idx >> 4;         // row 0..63
    int c4  = idx & 15;         // float4 column 0..15
    v4f v = *(const v4f*)(qb + (size_t)(tile_m + r) * Dd + c4 * 4);
    *(v4f*)(q_s + r * LDS_STRIDE + c4 * 4) = v;
  }
  // ---- stage k tile: 128 rows x 64 cols = 2048 float4, 8 per thread ----
#pragma unroll
  for (int i = 0; i < 8; ++i) {
    int idx = tid + i * 256;
    int r   = idx >> 4;         // row 0..127
    int c4  = idx & 15;
    v4f v = *(const v4f*)(kb + (size_t)(tile_n + r) * Dd + c4 * 4);
    *(v4f*)(k_s + r * LDS_STRIDE + c4 * 4) = v;
  }
  __syncthreads();

  // ---- per-row squared norms (from LDS), threads 0..191 ----
  if (tid < WG_M + WG_N) {
    const float* row = (tid < WG_M) ? (q_s + tid * LDS_STRIDE)
                                    : (k_s + (tid - WG_M) * LDS_STRIDE);
    float s = 0.0f;
#pragma unroll
    for (int d = 0; d < Dd; ++d) s = fmaf(row[d], row[d], s);
    if (tid < WG_M) qn_s[tid] = s;
    else            kn_s[tid - WG_M] = s;
  }
  __syncthreads();

  // ---- WMMA: each wave owns a 32x32 output tile (2x2 of 16x16) ----
  const int wave = tid >> 5;        // 0..7
  const int lane = tid & 31;
  const int half = lane >> 4;       // 0 or 1 (selects K pair / M+8)
  const int lm   = lane & 15;       // row-within-16
  const int wm   = (wave >> 2) * 32;   // 0 or 32
  const int wn   = (wave & 3)  * 32;   // 0,32,64,96

  v8f c00 = {}, c01 = {}, c10 = {}, c11 = {};

#pragma unroll
  for (int kk = 0; kk < Dd / 4; ++kk) {
    const int kb4 = kk * 4 + half * 2;   // this lane's K pair within the K=4 chunk
    // A fragments (q rows): lane l -> M = lm, VGPR0/1 = K kb4, kb4+1
    v2f a0 = *(const v2f*)(q_s + (wm + lm)      * LDS_STRIDE + kb4);
    v2f a1 = *(const v2f*)(q_s + (wm + 16 + lm) * LDS_STRIDE + kb4);
    // B fragments (k^T): B[d][n] = k[n][d] -> identical index math on k tile
    v2f b0 = *(const v2f*)(k_s + (wn + lm)      * LDS_STRIDE + kb4);
    v2f b1 = *(const v2f*)(k_s + (wn + 16 + lm) * LDS_STRIDE + kb4);

    c00 = __builtin_amdgcn_wmma_f32_16x16x4_f32(false, a0, false, b0, (short)0, c00, false, false);
    c01 = __builtin_amdgcn_wmma_f32_16x16x4_f32(false, a0, false, b1, (short)0, c01, false, false);
    c10 = __builtin_amdgcn_wmma_f32_16x16x4_f32(false, a1, false, b0, (short)0, c10, false, false);
    c11 = __builtin_amdgcn_wmma_f32_16x16x4_f32(false, a1, false, b1, (short)0, c11, false, false);
  }

  // ---- epilogue: hyperbolic distance + store ----
  float* __restrict__ ob = out + (size_t)bh * Nn * Nn;

  auto epilogue = [&](v8f acc, int mbase, int nbase) {
    const int n_local = nbase + lm;            // C layout: lane%16 = N
    const float kn_v  = kn_s[n_local];
    const float kd    = 1.0f - kn_v;
#pragma unroll
    for (int v = 0; v < 8; ++v) {
      const int m_local = mbase + v + half * 8; // C layout: VGPR v -> M = v (+8 for upper half)
      const float qn_v = qn_s[m_local];
      const float qk   = acc[v];
      float diff  = qn_v + kn_v - 2.0f * qk;
      diff        = fmaxf(diff, 0.0f);
      float denom = (1.0f - qn_v) * kd;
      denom       = fmaxf(denom, 1e-6f);
      float carg  = 1.0f + 2.0f * diff / denom;
      carg        = fmaxf(carg, 1.0f);
      // arccosh(x) = log(x + sqrt(x^2 - 1))
      const float d = __logf(carg + __fsqrt_rn(fmaf(carg, carg, -1.0f)));
      ob[(size_t)(tile_m + m_local) * Nn + (tile_n + n_local)] = d;
    }
  };

  epilogue(c00, wm,      wn);
  epilogue(c01, wm,      wn + 16);
  epilogue(c10, wm + 16, wn);
  epilogue(c11, wm + 16, wn + 16);
}

extern "C" void kernel_launch(void* const* d_in, const int* in_sizes, int n_in,
                              void* d_out, int out_size, void* d_ws, size_t ws_size,
                              hipStream_t stream) {
  (void)in_sizes; (void)n_in; (void)d_ws; (void)ws_size; (void)out_size;
  const float* q = (const float*)d_in[0];
  const float* k = (const float*)d_in[1];
  float* out = (float*)d_out;

  dim3 grid(Nn / WG_N, Nn / WG_M, Bb * Hh);  // (16, 32, 32)
  dim3 block(256, 1, 1);                     // 8 wave32s
  hyperdist_wmma_kernel<<<grid, block, 0, stream>>>(q, k, out);
}